// EventEncoder_19232863552116
// MI455X (gfx1250) — compile-verified
//
#include <hip/hip_runtime.h>

#define NUM_NODES 10000
#define BSZ 32
#define D 64
#define DH 256
#define EV 128                      // events per block (8 M-tiles of 16)
#define LN_EPS 1e-5f

typedef __attribute__((ext_vector_type(2))) float v2f;
typedef __attribute__((ext_vector_type(8))) float v8f;

// Dynamic LDS layout (bytes):
//   s_u   : EV ints
//   s_i   : EV ints
//   s_msg : EV floats
//   s_mu  : EV floats
//   s_rs  : EV floats
//   s_red : EV*8*2 floats
//   s_big : union{ [xu EV*65 | xi EV*65] , h EV*257 } floats
#define HDR_FLOATS   (5 * EV)
#define RED_FLOATS   (EV * 16)
#define XU_STRIDE    (D + 1)
#define H_STRIDE     (DH + 1)
#define BIG_FLOATS   (EV * H_STRIDE)          // 128*257 > 2*128*65
#define SMEM_BYTES   ((HDR_FLOATS + RED_FLOATS + BIG_FLOATS) * 4)

// ---------------------------------------------------------------------------
__global__ void zero_kernel(float* __restrict__ p, long n) {
  long i = (long)blockIdx.x * blockDim.x + threadIdx.x;
  if (i < n) p[i] = 0.0f;
}

// ---------------------------------------------------------------------------
// 256 threads = 8 waves. Wave w: ns = w&3 (64-col slab), mhalf = w>>2.
// Wave computes M-tiles {mhalf*4 .. mhalf*4+3} x N-tiles {ns*4 .. ns*4+3}
// -> 16 C-tiles (128 acc VGPRs), 4x B-fragment reuse across M.
// ---------------------------------------------------------------------------
__global__ __launch_bounds__(256) void enc_kernel(
    const int* __restrict__ users, const int* __restrict__ items,
    const float* __restrict__ messages, int E,
    const float* __restrict__ user_emb, const float* __restrict__ item_emb,
    const float* __restrict__ W1, const float* __restrict__ b1,
    const float* __restrict__ ln_g, const float* __restrict__ ln_b,
    const float* __restrict__ W2, const float* __restrict__ b2,
    float* __restrict__ out_u, float* __restrict__ out_i,
    float* __restrict__ ugsum, float* __restrict__ igsum)
{
  extern __shared__ char smem_raw[];
  int*   s_u   = (int*)smem_raw;
  int*   s_i   = s_u + EV;
  float* s_msg = (float*)(s_i + EV);
  float* s_mu  = s_msg + EV;
  float* s_rs  = s_mu + EV;
  float* s_red = s_rs + EV;                 // [EV][8][2]
  float* s_big = s_red + RED_FLOATS;
  float* s_xu  = s_big;                     // [EV][65]   (phase 1)
  float* s_xi  = s_big + EV * XU_STRIDE;    // [EV][65]   (phase 1)
  float* s_h   = s_big;                     // [EV][257]  (phase 2+)

  const int tid  = threadIdx.x;
  const int base = blockIdx.x * EV;

  // ---- event ids / messages ----------------------------------------------
  if (tid < EV) {
    int e = base + tid;
    s_u[tid]   = (e < E) ? users[e]    : 0;
    s_i[tid]   = (e < E) ? items[e]    : 0;
    s_msg[tid] = (e < E) ? messages[e] : 0.0f;
  }
  __syncthreads();

  // ---- gather embeddings (EV*64 each, coalesced rows) --------------------
  for (int it = 0; it < (EV * D) / 256; ++it) {
    int idx = it * 256 + tid;
    int ev = idx >> 6, d = idx & 63;
    s_xu[ev * XU_STRIDE + d] = user_emb[(s_u[ev] % NUM_NODES) * D + d];
    s_xi[ev * XU_STRIDE + d] = item_emb[(s_i[ev] % NUM_NODES) * D + d];
  }
  __syncthreads();

  // ---- WMMA decomposition ------------------------------------------------
  const int lane  = tid & 31;
  const int wave  = tid >> 5;
  const int mhalf = wave >> 2;            // 0..1
  const int ns    = wave & 3;             // 0..3
  const int lm    = lane & 15;
  const int khalf = (lane >> 4) << 1;     // 0 or 2
  const int mb0   = mhalf * 4;            // first M-tile index

  v8f acc[4][4] = {};                     // [m-tile][n-tile]

  // GEMM1a: ue @ W1[0:64, :]
  for (int k0 = 0; k0 < D; k0 += 4) {
    v2f a[4], b[4];
#pragma unroll
    for (int j = 0; j < 4; ++j) {
      const float* row = &s_xu[((mb0 + j) * 16 + lm) * XU_STRIDE + k0 + khalf];
      a[j].x = row[0]; a[j].y = row[1];
    }
#pragma unroll
    for (int t = 0; t < 4; ++t) {
      int n = ns * 64 + t * 16 + lm;
      b[t].x = W1[(k0 + khalf) * DH + n];
      b[t].y = W1[(k0 + khalf + 1) * DH + n];
    }
#pragma unroll
    for (int j = 0; j < 4; ++j)
#pragma unroll
      for (int t = 0; t < 4; ++t)
        acc[j][t] = __builtin_amdgcn_wmma_f32_16x16x4_f32(
            false, a[j], false, b[t], (short)0, acc[j][t], false, false);
  }
  // GEMM1b: ie @ W1[64:128, :]
  for (int k0 = 0; k0 < D; k0 += 4) {
    v2f a[4], b[4];
#pragma unroll
    for (int j = 0; j < 4; ++j) {
      const float* row = &s_xi[((mb0 + j) * 16 + lm) * XU_STRIDE + k0 + khalf];
      a[j].x = row[0]; a[j].y = row[1];
    }
#pragma unroll
    for (int t = 0; t < 4; ++t) {
      int n = ns * 64 + t * 16 + lm;
      b[t].x = W1[(64 + k0 + khalf) * DH + n];
      b[t].y = W1[(64 + k0 + khalf + 1) * DH + n];
    }
#pragma unroll
    for (int j = 0; j < 4; ++j)
#pragma unroll
      for (int t = 0; t < 4; ++t)
        acc[j][t] = __builtin_amdgcn_wmma_f32_16x16x4_f32(
            false, a[j], false, b[t], (short)0, acc[j][t], false, false);
  }

  // all xu/xi reads done -> safe to overlay h on the same LDS
  __syncthreads();

  // rank-1 message term + bias; spill h.
  // C layout: VGPR r, lane<16 -> M=r ; lane>=16 -> M=r+8 ; N = lane&15.
  {
#pragma unroll
    for (int t = 0; t < 4; ++t) {
      int n = ns * 64 + t * 16 + lm;
      float wl = W1[128 * DH + n];
      float bb = b1[n];
#pragma unroll
      for (int j = 0; j < 4; ++j) {
        int mbase = (mb0 + j) * 16 + ((lane >> 4) << 3);
#pragma unroll
        for (int r = 0; r < 8; ++r) {
          int m = mbase + r;
          s_h[m * H_STRIDE + n] = acc[j][t][r] + s_msg[m] * wl + bb;
        }
      }
    }
  }
  __syncthreads();

  // ---- LayerNorm: EV rows x 8 partials -----------------------------------
  for (int it = 0; it < (EV * 8) / 256; ++it) {
    int task = it * 256 + tid;
    int row = task >> 3, part = task & 7;
    float s = 0.0f, ss = 0.0f;
    const float* hr = &s_h[row * H_STRIDE + part * 32];
#pragma unroll 8
    for (int c = 0; c < 32; ++c) { float v = hr[c]; s += v; ss += v * v; }
    s_red[(row * 8 + part) * 2 + 0] = s;
    s_red[(row * 8 + part) * 2 + 1] = ss;
  }
  __syncthreads();
  if (tid < EV) {
    float s = 0.0f, ss = 0.0f;
    for (int p = 0; p < 8; ++p) {
      s  += s_red[(tid * 8 + p) * 2 + 0];
      ss += s_red[(tid * 8 + p) * 2 + 1];
    }
    float mu  = s * (1.0f / DH);
    float var = ss * (1.0f / DH) - mu * mu;
    s_mu[tid] = mu;
    s_rs[tid] = rsqrtf(var + LN_EPS);
  }
  __syncthreads();
  {
    float g = ln_g[tid], bb = ln_b[tid];   // each thread owns column `tid`
    for (int ev = 0; ev < EV; ++ev) {
      float* p = &s_h[ev * H_STRIDE + tid];
      *p = (*p - s_mu[ev]) * s_rs[ev] * g + bb;
    }
  }
  __syncthreads();

  // ---- GEMM2: hn @ W2 (K = 256) ------------------------------------------
#pragma unroll
  for (int j = 0; j < 4; ++j)
#pragma unroll
    for (int t = 0; t < 4; ++t) acc[j][t] = (v8f){};

  for (int k0 = 0; k0 < DH; k0 += 4) {
    v2f a[4], b[4];
#pragma unroll
    for (int j = 0; j < 4; ++j) {
      const float* row = &s_h[((mb0 + j) * 16 + lm) * H_STRIDE + k0 + khalf];
      a[j].x = row[0]; a[j].y = row[1];
    }
#pragma unroll
    for (int t = 0; t < 4; ++t) {
      int n = ns * 64 + t * 16 + lm;
      b[t].x = W2[(k0 + khalf) * DH + n];
      b[t].y = W2[(k0 + khalf + 1) * DH + n];
    }
#pragma unroll
    for (int j = 0; j < 4; ++j)
#pragma unroll
      for (int t = 0; t < 4; ++t)
        acc[j][t] = __builtin_amdgcn_wmma_f32_16x16x4_f32(
            false, a[j], false, b[t], (short)0, acc[j][t], false, false);
  }
  __syncthreads();   // hn reads complete -> reuse buffer for enc
  {
#pragma unroll
    for (int t = 0; t < 4; ++t) {
      int n = ns * 64 + t * 16 + lm;
      float bb = b2[n];
#pragma unroll
      for (int j = 0; j < 4; ++j) {
        int mbase = (mb0 + j) * 16 + ((lane >> 4) << 3);
#pragma unroll
        for (int r = 0; r < 8; ++r)
          s_h[(mbase + r) * H_STRIDE + n] = acc[j][t][r] + bb;
      }
    }
  }
  __syncthreads();

  // ---- scatter: column `tid` of each event, coalesced atomics ------------
  // enc cols: [0,64) uu -> out_u ; [64,128) iu -> out_i ;
  //           [128,192) ug -> ugsum ; [192,256) ig -> igsum
  const int col = tid;
  for (int ev = 0; ev < EV; ++ev) {
    if (base + ev >= E) break;
    float v = s_h[ev * H_STRIDE + col];
    int nu = s_u[ev], ni = s_i[ev];
    if (col < 64) {
      int b = nu % BSZ, nn = nu / BSZ;
      atomicAdd(&out_u[(long)b * NUM_NODES * D + (long)nn * D + col], v);
    } else if (col < 128) {
      int b = ni % BSZ, nn = ni / BSZ;
      atomicAdd(&out_i[(long)b * NUM_NODES * D + (long)nn * D + (col - 64)], v);
    } else if (col < 192) {
      atomicAdd(&ugsum[(nu % BSZ) * D + (col - 128)], v);
    } else {
      atomicAdd(&igsum[(ni % BSZ) * D + (col - 192)], v);
    }
  }
}

// ---------------------------------------------------------------------------
__global__ void finalize_kernel(float* __restrict__ out,
                                const float* __restrict__ ugsum,
                                const float* __restrict__ igsum) {
  const long half = (long)BSZ * NUM_NODES * D;
  long i = (long)blockIdx.x * blockDim.x + threadIdx.x;
  if (i >= 2 * half) return;
  const float inv = 1.0f / (float)NUM_NODES;
  if (i < half) {
    long b = i / ((long)NUM_NODES * D);
    int  d = (int)(i % D);
    out[i] += ugsum[b * D + d] * inv;
  } else {
    long j = i - half;
    long b = j / ((long)NUM_NODES * D);
    int  d = (int)(j % D);
    out[i] += igsum[b * D + d] * inv;
  }
}

// ---------------------------------------------------------------------------
extern "C" void kernel_launch(void* const* d_in, const int* in_sizes, int n_in,
                              void* d_out, int out_size, void* d_ws, size_t ws_size,
                              hipStream_t stream) {
  const int*   users    = (const int*)d_in[0];
  const int*   items    = (const int*)d_in[1];
  const float* messages = (const float*)d_in[2];
  // d_in[3] = bs scalar (compile-time 32 here)
  const float* user_emb = (const float*)d_in[4];
  const float* item_emb = (const float*)d_in[5];
  const float* W1       = (const float*)d_in[6];
  const float* b1       = (const float*)d_in[7];
  const float* ln_g     = (const float*)d_in[8];
  const float* ln_b     = (const float*)d_in[9];
  const float* W2       = (const float*)d_in[10];
  const float* b2       = (const float*)d_in[11];

  const int E = in_sizes[0];
  float* out = (float*)d_out;
  const long half = (long)BSZ * NUM_NODES * D;  // 20,480,000
  float* out_u = out;
  float* out_i = out + half;
  float* ugsum = (float*)d_ws;                  // [BSZ][D]
  float* igsum = ugsum + BSZ * D;               // [BSZ][D]

  const long tot = 2 * half;
  zero_kernel<<<(int)((tot + 1023) / 1024), 1024, 0, stream>>>(out, tot);
  zero_kernel<<<(2 * BSZ * D + 255) / 256, 256, 0, stream>>>(ugsum, 2 * BSZ * D);

  const int nblk = (E + EV - 1) / EV;
  enc_kernel<<<nblk, 256, SMEM_BYTES, stream>>>(
      users, items, messages, E, user_emb, item_emb, W1, b1, ln_g, ln_b,
      W2, b2, out_u, out_i, ugsum, igsum);

  finalize_kernel<<<(int)((tot + 255) / 256), 256, 0, stream>>>(out, ugsum, igsum);
}